// Encoder_73091753443852
// MI455X (gfx1250) — compile-verified
//
#include <hip/hip_runtime.h>
#include <cmath>

// ---------------------------------------------------------------------------
// Bidirectional 2-layer LSTM encoder for MI455X (gfx1250, wave32).
//
// Roofline: ~21.5 GFLOP fp32 total, ~50 MB of traffic, 10.5 MB of weights
// (L2-resident, L2 = 192 MB). At 23.3 TB/s HBM this is latency-bound on the
// 4096-step recurrence, NOT tensor-throughput bound. Therefore:
//   * input projections (x @ Wih^T, all timesteps at once) -> fp32 WMMA GEMM
//     (V_WMMA_F32_16X16X4_F32), keeping full reference precision,
//   * the per-step h @ Whh^T (256 -> 1024 GEMV) stays VALU FMA: a WMMA tile
//     would use 1/16 of its M dimension for a GEMV.
// ---------------------------------------------------------------------------

#define TT   4096
#define INW  512
#define HW   256
#define GW   1024   /* 4*H gate width */

typedef __attribute__((ext_vector_type(2))) float v2f;
typedef __attribute__((ext_vector_type(8))) float v8f;

__device__ __forceinline__ float sigm(float x) { return 1.0f / (1.0f + expf(-x)); }

// ---------------------------------------------------------------------------
// 1) Embedding gather: x[t][:] = emb[tokens[t]][:]   (T x 512, float4 copies)
// ---------------------------------------------------------------------------
__global__ __launch_bounds__(128) void gather_embed(const int* __restrict__ tokens,
                                                    const float* __restrict__ emb,
                                                    float* __restrict__ x) {
    const int t = blockIdx.x;
    const int tok = tokens[t];
    const float4* src = reinterpret_cast<const float4*>(emb + (size_t)tok * INW);
    float4* dst = reinterpret_cast<float4*>(x + (size_t)t * INW);
    dst[threadIdx.x] = src[threadIdx.x];   // 128 threads * float4 = 512 floats
}

// ---------------------------------------------------------------------------
// 2) Input projection GEMM:  G[d][s][j] = sum_k X_d[row(s)][k] * W_d[j][k] + bias[j]
//    One wave (wave32) per 16x16 output tile, K-loop of V_WMMA_F32_16X16X4_F32.
//    revB!=0 -> backward direction reads X rows reversed (layer 0 only).
//
//    Fragment layouts (CDNA5 ISA 7.12.2, 32-bit):
//      A 16x4:  lane<16: M=lane,    holds K=k0 (v0), k0+1 (v1)
//               lane>=16: M=lane-16, holds K=k0+2 (v0), k0+3 (v1)
//      B 4x16 (mirrored): lane gives N=lane&15, K-half split like A.
//      C 16x16: VGPR r -> M = r + (lane>=16 ? 8 : 0), N = lane&15.
// ---------------------------------------------------------------------------
__global__ __launch_bounds__(256) void gemm_wmma_f32(
        const float* __restrict__ XF, const float* __restrict__ XB,
        const float* __restrict__ WF, const float* __restrict__ WB,
        const float* __restrict__ bihF, const float* __restrict__ bhhF,
        const float* __restrict__ bihB, const float* __restrict__ bhhB,
        float* __restrict__ Gout, int K, int revB) {
    const int d    = blockIdx.y;                       // 0 = forward, 1 = backward
    const float* X  = d ? XB : XF;
    const float* W  = d ? WB : WF;
    const float* bi = d ? bihB : bihF;
    const float* bh = d ? bhhB : bhhF;

    const int wave = threadIdx.x >> 5;
    const int lane = threadIdx.x & 31;
    const int tile = blockIdx.x * 8 + wave;            // 8 waves per block
    const int tn   = tile & 63;                        // 1024/16 = 64 N-tiles
    const int tm   = tile >> 6;                        // 4096/16 = 256 M-tiles
    const int m0   = tm * 16;
    const int n0   = tn * 16;

    const int half = lane >> 4;                        // K-half select (0/1)
    const int l15  = lane & 15;
    const int srow = m0 + l15;                         // step index of this A-row
    const int xrow = (revB && d) ? (TT - 1 - srow) : srow;

    const float* arow = X + (size_t)xrow * K;
    const float* brow = W + (size_t)(n0 + l15) * K;    // W is (1024 x K); B = W^T tile
    const int koff = half * 2;

    v8f c = {};
    for (int k0 = 0; k0 < K; k0 += 4) {
        v2f a, b;
        a.x = arow[k0 + koff];
        a.y = arow[k0 + koff + 1];
        b.x = brow[k0 + koff];
        b.y = brow[k0 + koff + 1];
        // D = A(16x4) * B(4x16) + C   -- v_wmma_f32_16x16x4_f32
        c = __builtin_amdgcn_wmma_f32_16x16x4_f32(false, a, false, b,
                                                  (short)0, c, false, false);
    }

    const int col = n0 + l15;
    const float bias = bi[col] + bh[col];
    float* Gd = Gout + (size_t)d * TT * GW;
#pragma unroll
    for (int r = 0; r < 8; ++r) {
        const int row = m0 + r + half * 8;             // C-fragment M mapping
        Gd[(size_t)row * GW + col] = c[r] + bias;
    }
}

// ---------------------------------------------------------------------------
// 3) Recurrence. grid = 2 (direction), block = 1024 (one WGP: 32 wave32s).
//    Thread j owns gate row j: g[j] = G[t][j] + dot(Whh[j][:], h[:]).
//    h, c, gates live in LDS; Whh streams from L2 (1 MB/step, trivial).
//    Two barriers per step: (gates ready) and (h/c updated).
// ---------------------------------------------------------------------------
__global__ __launch_bounds__(1024) void lstm_recur(
        const float* __restrict__ G,
        const float* __restrict__ WhhF, const float* __restrict__ WhhB,
        float* __restrict__ hseq,            // layer 0: [2][T][256] out
        float* __restrict__ dout,            // full d_out base
        int layer) {
    __shared__ __align__(16) float hsm[HW];
    __shared__ float csm[HW];
    __shared__ float gsm[GW];

    const int d   = blockIdx.x;
    const int tid = threadIdx.x;
    const float*  W    = d ? WhhB : WhhF;
    const float4* wrow = reinterpret_cast<const float4*>(W + (size_t)tid * HW);
    const float*  Gd   = G + (size_t)d * TT * GW;
    float* outSeq = dout + 2048;             // after cell(1024)+hidden(1024)

    if (tid < HW) { hsm[tid] = 0.0f; csm[tid] = 0.0f; }
    __syncthreads();

    const float4* h4 = reinterpret_cast<const float4*>(hsm);

    for (int t = 0; t < TT; ++t) {
        float acc = Gd[(size_t)t * GW + tid];
        float a0 = 0.f, a1 = 0.f, a2 = 0.f, a3 = 0.f;
#pragma unroll 8
        for (int k = 0; k < HW / 4; ++k) {
            const float4 w  = wrow[k];
            const float4 hv = h4[k];         // same-address LDS broadcast
            a0 = fmaf(w.x, hv.x, a0);
            a1 = fmaf(w.y, hv.y, a1);
            a2 = fmaf(w.z, hv.z, a2);
            a3 = fmaf(w.w, hv.w, a3);
        }
        gsm[tid] = acc + ((a0 + a1) + (a2 + a3));
        __syncthreads();                     // gates visible; old-h reads done

        if (tid < HW) {
            const float i_ = gsm[tid];
            const float f_ = gsm[tid + HW];
            const float g_ = gsm[tid + 2 * HW];
            const float o_ = gsm[tid + 3 * HW];
            const float cNew = sigm(f_) * csm[tid] + sigm(i_) * tanhf(g_);
            const float hNew = sigm(o_) * tanhf(cNew);
            csm[tid] = cNew;
            hsm[tid] = hNew;
            if (layer == 0) {
                hseq[((size_t)d * TT + t) * HW + tid] = hNew;   // step order
            } else {
                if (d == 0) outSeq[(size_t)t * (2 * HW) + tid] = hNew;
                else        outSeq[(size_t)(TT - 1 - t) * (2 * HW) + HW + tid] = hNew;
            }
        }
        __syncthreads();                     // new h visible for next step
    }

    if (tid < HW) {                          // finals: (L,2,H) cell then hidden
        dout[(layer * 2 + d) * HW + tid]        = csm[tid];
        dout[1024 + (layer * 2 + d) * HW + tid] = hsm[tid];
    }
}

// ---------------------------------------------------------------------------
// Host-side orchestration (all on `stream`, graph-capture safe).
// Workspace layout (floats):
//   x    : TT*INW                 =  2,097,152   ( 8 MB)
//   Gbuf : 2*TT*GW                =  8,388,608   (32 MB)  (reused L0 -> L1)
//   hs   : 2*TT*HW                =  2,097,152   ( 8 MB)
// ---------------------------------------------------------------------------
extern "C" void kernel_launch(void* const* d_in, const int* in_sizes, int n_in,
                              void* d_out, int out_size, void* d_ws, size_t ws_size,
                              hipStream_t stream) {
    const int*   tokens = (const int*)  d_in[0];
    const float* emb    = (const float*)d_in[1];
    const float* fWih0  = (const float*)d_in[2];
    const float* fWhh0  = (const float*)d_in[3];
    const float* fbih0  = (const float*)d_in[4];
    const float* fbhh0  = (const float*)d_in[5];
    const float* fWih1  = (const float*)d_in[6];
    const float* fWhh1  = (const float*)d_in[7];
    const float* fbih1  = (const float*)d_in[8];
    const float* fbhh1  = (const float*)d_in[9];
    const float* bWih0  = (const float*)d_in[10];
    const float* bWhh0  = (const float*)d_in[11];
    const float* bbih0  = (const float*)d_in[12];
    const float* bbhh0  = (const float*)d_in[13];
    const float* bWih1  = (const float*)d_in[14];
    const float* bWhh1  = (const float*)d_in[15];
    const float* bbih1  = (const float*)d_in[16];
    const float* bbhh1  = (const float*)d_in[17];

    float* out = (float*)d_out;
    float* wsf = (float*)d_ws;
    float* x    = wsf;
    float* Gbuf = x + (size_t)TT * INW;
    float* hs   = Gbuf + (size_t)2 * TT * GW;

    // 1) embedding gather
    gather_embed<<<TT, 128, 0, stream>>>(tokens, emb, x);

    // 2) layer-0 input projections (K=512; backward reads x reversed)
    {
        dim3 grid((TT / 16) * (GW / 16) / 8, 2);
        gemm_wmma_f32<<<grid, 256, 0, stream>>>(x, x, fWih0, bWih0,
                                                fbih0, fbhh0, bbih0, bbhh0,
                                                Gbuf, INW, /*revB=*/1);
    }

    // 3) layer-0 recurrence -> hs (step order), finals l=0
    lstm_recur<<<2, 1024, 0, stream>>>(Gbuf, fWhh0, bWhh0, hs, out, /*layer=*/0);

    // 4) layer-1 input projections (K=256; inputs already in step order)
    {
        dim3 grid((TT / 16) * (GW / 16) / 8, 2);
        gemm_wmma_f32<<<grid, 256, 0, stream>>>(hs, hs + (size_t)TT * HW,
                                                fWih1, bWih1,
                                                fbih1, fbhh1, bbih1, bbhh1,
                                                Gbuf, HW, /*revB=*/0);
    }

    // 5) layer-1 recurrence -> d_out outputs region + finals l=1
    lstm_recur<<<2, 1024, 0, stream>>>(Gbuf, fWhh1, bWhh1, hs, out, /*layer=*/1);
}